// HyperbolicMemory_74663711474149
// MI455X (gfx1250) — compile-verified
//
#include <hip/hip_runtime.h>
#include <math.h>

typedef __attribute__((ext_vector_type(2))) float v2f;
typedef __attribute__((ext_vector_type(8))) float v8f;

#define B_ROWS 2048
#define N_MEM  100000
#define DIM    128
#define TOPK   16
#define TEMPR  0.1f
#define MAXR   0.9f

#define NT      2048      // columns of the distance tile per block
#define NTILES  49        // ceil(N_MEM / NT)
#define QSTRIDE 132       // padded LDS row stride for q tile (bank-conflict-free A reads)
#define DSTRIDE 2052      // padded LDS row stride for dist tile
#define NCAND   (NTILES * TOPK)   // 784 candidates per query row
#define FLT_BIG 3.0e38f

// ---------------------------------------------------------------------------
// Phase 0: h = tanh(x @ W^T + b), rescale into Poincare ball radius 0.9.
// One block (128 threads) per row; also emits ||h||^2.
// ---------------------------------------------------------------------------
__global__ void __launch_bounds__(DIM)
proj_kernel(const float* __restrict__ src, const float* __restrict__ W,
            const float* __restrict__ bvec, float* __restrict__ dst,
            float* __restrict__ dst2) {
    __shared__ float x[DIM];
    __shared__ float red[DIM];
    const int row = blockIdx.x;
    const int tid = threadIdx.x;

    x[tid] = src[(size_t)row * DIM + tid];
    __syncthreads();

    float acc = bvec[tid];
    const float* wr = W + (size_t)tid * DIM;   // out[j] = sum_d x[d] * W[j][d]
#pragma unroll 8
    for (int d = 0; d < DIM; ++d) acc = fmaf(x[d], wr[d], acc);

    float h = tanhf(acc);
    red[tid] = h * h;
    __syncthreads();
    for (int s = DIM / 2; s > 0; s >>= 1) {
        if (tid < s) red[tid] += red[tid + s];
        __syncthreads();
    }
    const float n2   = red[0];
    float norm       = fmaxf(sqrtf(n2), 1e-8f);
    const float scl  = (norm > MAXR) ? (MAXR / norm) : 1.0f;
    h *= scl;
    dst[(size_t)row * DIM + tid] = h;
    if (tid == 0) dst2[row] = n2 * scl * scl;
}

// ---------------------------------------------------------------------------
// Phase 1: fused distance GEMM (fp32 WMMA) + per-tile top-16.
// grid = (B/16, NTILES); block = 256 (8 waves).
// Wave w computes 16 contiguous 16x16 output tiles of the 16xNT distance tile.
// ---------------------------------------------------------------------------
__global__ void __launch_bounds__(256)
dist_topk_kernel(const float* __restrict__ q,  const float* __restrict__ q2,
                 const float* __restrict__ mv, const float* __restrict__ m2,
                 float* __restrict__ cand_d,   int* __restrict__ cand_i) {
    extern __shared__ float smem[];
    float* q_l  = smem;                        // 16 * QSTRIDE
    float* q2_l = smem + 16 * QSTRIDE;         // 16
    float* dist = q2_l + 16;                   // 16 * DSTRIDE

    const int tid   = threadIdx.x;
    const int lane  = tid & 31;
    const int wave  = tid >> 5;
    const int qbase = blockIdx.x * 16;
    const int cbase = blockIdx.y * NT;

    // Stage q tile (16 x 128) and q2 into LDS.
    for (int t = tid; t < 16 * DIM; t += 256) {
        const int r = t >> 7, c = t & 127;
        q_l[r * QSTRIDE + c] = q[(size_t)(qbase + r) * DIM + c];
    }
    if (tid < 16) q2_l[tid] = q2[qbase + tid];
    __syncthreads();

    // WMMA fragment lane mapping (V_WMMA_F32_16X16X4_F32):
    //  A (16x4): lane holds row m = lane&15, K = (lane>>4)*2 + {0,1}
    //  B (4x16): lane holds col n = lane&15, K = (lane>>4)*2 + {0,1}
    //  C (16x16): VGPR r -> row r + 8*(lane>>4), col = lane&15
    const int mrow  = lane & 15;
    const int khalf = (lane >> 4) << 1;        // 0 or 2

    for (int ct = wave * 16; ct < wave * 16 + 16; ++ct) {
        const int col_l0 = ct * 16;
        const int n  = cbase + col_l0 + (lane & 15);
        const int nc = (n < N_MEM) ? n : (N_MEM - 1);
        const float* mp = mv + (size_t)nc * DIM;

        v8f acc = {0.f, 0.f, 0.f, 0.f, 0.f, 0.f, 0.f, 0.f};
#pragma unroll 4
        for (int kb = 0; kb < DIM; kb += 4) {
            const int koff = kb + khalf;
            v2f a  = *(const v2f*)(q_l + mrow * QSTRIDE + koff);  // ds_load_b64
            v2f bb = *(const v2f*)(mp + koff);                    // global_load_b64 (L2 hit)
            acc = __builtin_amdgcn_wmma_f32_16x16x4_f32(
                false, a, false, bb, (short)0, acc, false, false);
        }

        const float m2v  = (n < N_MEM) ? m2[n] : 0.f;
        const int   rb   = (lane >> 4) << 3;   // 0 or 8
#pragma unroll
        for (int r = 0; r < 8; ++r) {
            const int row = rb + r;
            const float d2 = q2_l[row] + m2v - 2.0f * acc[r];
            const float dv = (n < N_MEM) ? sqrtf(fmaxf(d2, 1e-12f)) : FLT_BIG;
            dist[row * DSTRIDE + col_l0 + (lane & 15)] = dv;
        }
    }
    __syncthreads();

    // Per-tile top-16 per row: wave w owns rows w and w+8.
    for (int rr = 0; rr < 2; ++rr) {
        const int row = wave + rr * 8;
        float* drow = dist + row * DSTRIDE;
        const int growl = qbase + row;
        for (int it = 0; it < TOPK; ++it) {
            float best = FLT_BIG; int bestj = 0x7fffffff;
            for (int j = lane; j < NT; j += 32) {
                const float v = drow[j];
                if (v < best || (v == best && j < bestj)) { best = v; bestj = j; }
            }
            for (int off = 16; off > 0; off >>= 1) {
                const float ov = __shfl_xor(best, off, 32);
                const int   oj = __shfl_xor(bestj, off, 32);
                if (ov < best || (ov == best && oj < bestj)) { best = ov; bestj = oj; }
            }
            if (lane == 0) {
                const size_t base = ((size_t)growl * NTILES + blockIdx.y) * TOPK + it;
                cand_d[base] = best;
                cand_i[base] = cbase + bestj;
                drow[bestj]  = FLT_BIG;   // mask chosen entry
            }
            __syncthreads();
        }
    }
}

// ---------------------------------------------------------------------------
// Phase 2: merge 49x16 candidates per row -> final top-16, softmax, gather.
// One wave (32 threads) per query row.
// ---------------------------------------------------------------------------
__global__ void __launch_bounds__(32)
finalize_kernel(const float* __restrict__ cand_d, const int* __restrict__ cand_i,
                const float* __restrict__ mem_out, float* __restrict__ out) {
    __shared__ float cd[NCAND];
    __shared__ int   ci[NCAND];
    __shared__ float td[TOPK];
    __shared__ int   ti[TOPK];

    const int row  = blockIdx.x;
    const int lane = threadIdx.x;

    for (int j = lane; j < NCAND; j += 32) {
        cd[j] = cand_d[(size_t)row * NCAND + j];
        ci[j] = cand_i[(size_t)row * NCAND + j];
    }
    __syncthreads();

    for (int it = 0; it < TOPK; ++it) {
        float best = FLT_BIG; int bestj = 0x7fffffff;
        for (int j = lane; j < NCAND; j += 32) {
            const float v = cd[j];
            if (v < best || (v == best && j < bestj)) { best = v; bestj = j; }
        }
        for (int off = 16; off > 0; off >>= 1) {
            const float ov = __shfl_xor(best, off, 32);
            const int   oj = __shfl_xor(bestj, off, 32);
            if (ov < best || (ov == best && oj < bestj)) { best = ov; bestj = oj; }
        }
        if (lane == 0) { td[it] = best; ti[it] = ci[bestj]; cd[bestj] = FLT_BIG; }
        __syncthreads();
    }

    // Stabilized softmax(-d / TEMP); td[] is ascending so td[0] is the max logit.
    const float dmin = td[0];
    float sum = 0.f;
#pragma unroll
    for (int j = 0; j < TOPK; ++j) sum += expf(-(td[j] - dmin) / TEMPR);
    if (lane < TOPK) {
        out[(size_t)row * TOPK + lane] = expf(-(td[lane] - dmin) / TEMPR) / sum;
    }

    // Gather outcomes: out[B*K + (row*K + j)*D + f] = mem_out[idx_j * D + f]
    const size_t WOFF = (size_t)B_ROWS * TOPK;
    for (int t = lane; t < TOPK * DIM; t += 32) {
        const int j = t >> 7, f = t & 127;
        out[WOFF + ((size_t)row * TOPK + j) * DIM + f] =
            mem_out[(size_t)ti[j] * DIM + f];
    }
}

// ---------------------------------------------------------------------------
extern "C" void kernel_launch(void* const* d_in, const int* in_sizes, int n_in,
                              void* d_out, int out_size, void* d_ws, size_t ws_size,
                              hipStream_t stream) {
    (void)in_sizes; (void)n_in; (void)out_size; (void)ws_size;
    const float* query   = (const float*)d_in[0];
    const float* W       = (const float*)d_in[1];
    const float* bvec    = (const float*)d_in[2];
    const float* mem_emb = (const float*)d_in[3];
    const float* mem_out = (const float*)d_in[4];
    // d_in[5] = k (== 16, baked into TOPK)

    float* ws  = (float*)d_ws;
    float* q   = ws;                                   // 2048*128
    float* q2  = q  + (size_t)B_ROWS * DIM;            // 2048
    float* m   = q2 + B_ROWS;                          // 100000*128
    float* m2  = m  + (size_t)N_MEM * DIM;             // 100000
    float* cd  = m2 + N_MEM;                           // 2048*784
    int*   ci  = (int*)(cd + (size_t)B_ROWS * NCAND);  // 2048*784

    proj_kernel<<<B_ROWS, DIM, 0, stream>>>(query,   W, bvec, q, q2);
    proj_kernel<<<N_MEM,  DIM, 0, stream>>>(mem_emb, W, bvec, m, m2);

    dim3 grid(B_ROWS / 16, NTILES);
    const size_t shmem = (size_t)(16 * QSTRIDE + 16 + 16 * DSTRIDE) * sizeof(float);
    dist_topk_kernel<<<grid, 256, shmem, stream>>>(q, q2, m, m2, cd, ci);

    finalize_kernel<<<B_ROWS, 32, 0, stream>>>(cd, ci, mem_out, (float*)d_out);
}